// InternalGraphConvolutionLayer_63917703299187
// MI455X (gfx1250) — compile-verified
//
#include <hip/hip_runtime.h>

typedef __attribute__((ext_vector_type(2))) float v2f;
typedef __attribute__((ext_vector_type(8))) float v8f;

// ---------------------------------------------------------------------------
// Kernel 1: zero the workspace (agg[N*3] + s_accum[4])
// ---------------------------------------------------------------------------
__global__ void k_zero(float* __restrict__ p, int n) {
  int i = blockIdx.x * blockDim.x + threadIdx.x;
  if (i < n) p[i] = 0.0f;
}

// ---------------------------------------------------------------------------
// Kernel 2: edge scatter-add  agg[dst] += x[src]
// One thread per edge; 75M non-returning global_atomic_add_f32 that land in
// the 192MB L2 (agg is only 12MB), index streams are the HBM traffic.
// ---------------------------------------------------------------------------
__global__ __launch_bounds__(256) void k_edge(const float* __restrict__ x,
                                              const int*   __restrict__ src,
                                              const int*   __restrict__ dst,
                                              float*       __restrict__ agg,
                                              int E) {
  int e = blockIdx.x * blockDim.x + threadIdx.x;
  if (e >= E) return;
  int s = src[e] * 3;
  int d = dst[e] * 3;
  float x0 = x[s + 0];
  float x1 = x[s + 1];
  float x2 = x[s + 2];
  atomicAdd(agg + d + 0, x0);
  atomicAdd(agg + d + 1, x1);
  atomicAdd(agg + d + 2, x2);
}

// ---------------------------------------------------------------------------
// Kernel 3: node phase via V_WMMA_F32_16X16X4_F32 (CDNA5 shape, full f32).
//   h_pre = x @ W + agg @ M   for 16 nodes per wave per tile, then
//   s[j] += sum_nodes relu(h_pre[:, j]) for j in 0..2.
//
// A layout (16x4 f32, 2 VGPRs/lane): lane row M = lane&15;
//   lanes 0-15 hold (K0,K1), lanes 16-31 hold (K2,K3). We pad K3 = 0.
// B layout (4x16 f32, 2 VGPRs/lane): lane col N = lane&15, same K striping;
//   columns >= 3 are zero so the unused D columns are exactly zero.
// D layout (16x16 f32, 8 VGPRs/lane): lane holds column (lane&15),
//   VGPR r holds row r + 8*(lane>=16).
// WMMAs execute under full EXEC (all divergent guards reconverge before them).
// ---------------------------------------------------------------------------
__global__ __launch_bounds__(256) void k_node(const float* __restrict__ x,
                                              const float* __restrict__ agg,
                                              const float* __restrict__ W,
                                              const float* __restrict__ M,
                                              float*       __restrict__ s_out,
                                              int N) {
  const int lane  = threadIdx.x & 31;
  const int col   = lane & 15;          // column of B/D, row of A loaded by this lane
  const int hi    = lane >> 4;          // 0: K0/K1 half, 1: K2/K3 half
  const int wave  = blockIdx.x * (blockDim.x >> 5) + (threadIdx.x >> 5);
  const int nwav  = gridDim.x * (blockDim.x >> 5);

  // Broadcast W and M into the B-operand layout (padded 4x16).
  v2f bW = {0.0f, 0.0f};
  v2f bM = {0.0f, 0.0f};
  if (col < 3) {
    bW.x = hi ? W[2 * 3 + col] : W[0 * 3 + col];
    bW.y = hi ? 0.0f           : W[1 * 3 + col];
    bM.x = hi ? M[2 * 3 + col] : M[0 * 3 + col];
    bM.y = hi ? 0.0f           : M[1 * 3 + col];
  }

  float acc = 0.0f;                      // partial sum for column `col` (if col<3)
  const int ntiles = (N + 15) >> 4;

  for (int t = wave; t < ntiles; t += nwav) {
    const int n = t * 16 + col;          // node = A-row handled by this lane
    v2f aX = {0.0f, 0.0f};
    v2f aG = {0.0f, 0.0f};
    if (n < N) {
      const int base = 3 * n;
      aX.x = hi ? x[base + 2]   : x[base + 0];
      aX.y = hi ? 0.0f          : x[base + 1];
      aG.x = hi ? agg[base + 2] : agg[base + 0];
      aG.y = hi ? 0.0f          : agg[base + 1];
    }

    v8f c = {};
    // c = aX * W + c ; c = aG * M + c   (D = A x B + C, f32 throughout)
    c = __builtin_amdgcn_wmma_f32_16x16x4_f32(false, aX, false, bW,
                                              (short)0, c, false, false);
    c = __builtin_amdgcn_wmma_f32_16x16x4_f32(false, aG, false, bM,
                                              (short)0, c, false, false);

    if (col < 3) {
#pragma unroll
      for (int r = 0; r < 8; ++r) {
        const int node = t * 16 + r + (hi ? 8 : 0);
        const float v  = c[r];
        if (node < N) acc += (v > 0.0f) ? v : 0.0f;
      }
    }
  }

  if (col < 3) atomicAdd(s_out + col, acc);
}

// ---------------------------------------------------------------------------
// Kernel 4: softmax over the 3-element readout.
// ---------------------------------------------------------------------------
__global__ void k_softmax(const float* __restrict__ s_in, float* __restrict__ out) {
  if (threadIdx.x == 0 && blockIdx.x == 0) {
    const float a = s_in[0], b = s_in[1], c = s_in[2];
    const float m  = fmaxf(a, fmaxf(b, c));
    const float ea = __expf(a - m);
    const float eb = __expf(b - m);
    const float ec = __expf(c - m);
    const float inv = 1.0f / (ea + eb + ec);
    out[0] = ea * inv;
    out[1] = eb * inv;
    out[2] = ec * inv;
  }
}

// ---------------------------------------------------------------------------
// Launch
// ---------------------------------------------------------------------------
extern "C" void kernel_launch(void* const* d_in, const int* in_sizes, int n_in,
                              void* d_out, int out_size, void* d_ws, size_t ws_size,
                              hipStream_t stream) {
  (void)n_in; (void)out_size; (void)ws_size;

  const float* x   = (const float*)d_in[0];   // [N,3]
  const float* W   = (const float*)d_in[1];   // [3,3]
  const float* M   = (const float*)d_in[2];   // [3,3]
  const int*   src = (const int*)d_in[3];     // [E]
  const int*   dst = (const int*)d_in[4];     // [E]
  float*       out = (float*)d_out;           // [3]

  const int N = in_sizes[0] / 3;
  const int E = in_sizes[3];

  float* agg  = (float*)d_ws;                 // N*3 floats
  float* ssum = agg + (size_t)N * 3;          // 3 floats (+1 pad)

  // 1) zero agg + s accumulator (workspace is poisoned, not re-zeroed by harness)
  {
    const int n = N * 3 + 4;
    k_zero<<<(n + 255) / 256, 256, 0, stream>>>(agg, n);
  }

  // 2) edge scatter-add
  k_edge<<<(E + 255) / 256, 256, 0, stream>>>(x, src, dst, agg, E);

  // 3) WMMA node phase + global column sums
  {
    const int ntiles = (N + 15) / 16;
    int blocks = (ntiles + 7) / 8;            // 8 waves/block, 1 tile/wave/iter
    if (blocks > 16384) blocks = 16384;
    if (blocks < 1) blocks = 1;
    k_node<<<blocks, 256, 0, stream>>>(x, agg, W, M, ssum, N);
  }

  // 4) softmax readout
  k_softmax<<<1, 32, 0, stream>>>(ssum, out);
}